// GraphSAGE_63522566308230
// MI455X (gfx1250) — compile-verified
//
#include <hip/hip_runtime.h>
#include <hip/hip_bf16.h>
#include <cstdint>

// GraphSAGE (2x SAGEConv mean-aggregate) for MI455X / gfx1250.
// fp32 throughout; matrix ops use V_WMMA_F32_16X16X4_F32 (true fp32 WMMA).

#define N_NODES 50000
#define N_EDGES 800000
#define F_IN    128
#define F_HID   256
#define F_OUT   128

typedef __attribute__((ext_vector_type(2))) float v2f;
typedef __attribute__((ext_vector_type(8))) float v8f;

// ---------------------------------------------------------------- zero fill
__global__ __launch_bounds__(256) void zero_f32(float4* __restrict__ p, long nchunks) {
  long i = (long)blockIdx.x * blockDim.x + threadIdx.x;
  long s = (long)gridDim.x * blockDim.x;
  for (; i < nchunks; i += s) p[i] = make_float4(0.f, 0.f, 0.f, 0.f);
}

// ------------------------------------------------- edge scatter (mean prep)
// Consecutive threads cover consecutive float4 chunks of one edge's feature
// row -> coalesced 512B/1KB gathers; atomics land in L2 (agg fits in 192MB L2).
template <int F, bool DO_DEG>
__global__ __launch_bounds__(256) void scatter_accum(const float* __restrict__ feat,
                                                     const int* __restrict__ ei,
                                                     float* __restrict__ agg,
                                                     float* __restrict__ deg) {
  constexpr int CPE = F >> 2;  // float4 chunks per edge
  long t = (long)blockIdx.x * blockDim.x + threadIdx.x;
  int e  = (int)(t / CPE);
  int ch = (int)(t % CPE);
  if (e >= N_EDGES) return;
  int s = ei[e];
  int d = ei[N_EDGES + e];
  const float4 v = *(const float4*)(feat + (size_t)s * F + ch * 4);
  float* dp = agg + (size_t)d * F + ch * 4;
  unsafeAtomicAdd(dp + 0, v.x);
  unsafeAtomicAdd(dp + 1, v.y);
  unsafeAtomicAdd(dp + 2, v.z);
  unsafeAtomicAdd(dp + 3, v.w);
  if (DO_DEG && ch == 0) unsafeAtomicAdd(deg + d, 1.0f);
}

// --------------------------------------------------------- fused SAGE GEMM
// out[16 rows x NCOL] = act( (agg/max(deg,1)) @ Wl + xr @ Wr + bias )
// One block = 16 output rows; A tiles staged in LDS (padded vs bank conflicts);
// 8 waves, each wave owns 16-wide column tiles; fp32 WMMA 16x16x4.
template <int K, int NCOL, bool RELU>
__global__ __launch_bounds__(256) void sage_gemm(const float* __restrict__ agg,
                                                 const float* __restrict__ deg,
                                                 const float* __restrict__ xr,
                                                 const float* __restrict__ Wl,
                                                 const float* __restrict__ Wr,
                                                 const float* __restrict__ bias,
                                                 float* __restrict__ out) {
  constexpr int KP = K + 4;  // pad: row stride 4 banks apart -> conflict-free
  __shared__ float Am[16][KP];
  __shared__ float Ax[16][KP];

  const int row0 = blockIdx.x * 16;
  const int tid  = threadIdx.x;

  for (int i = tid; i < 16 * K; i += 256) {
    int r = i / K, c = i - r * K;
    float dg = deg[row0 + r];
    dg = dg > 1.0f ? dg : 1.0f;
    Am[r][c] = agg[(size_t)(row0 + r) * K + c] / dg;   // mean aggregation
    Ax[r][c] = xr[(size_t)(row0 + r) * K + c];
  }
  __syncthreads();

  const int lane = tid & 31;
  const int wave = tid >> 5;
  const int lh   = lane & 15;   // M (for A) / N (for B,C)
  const int hi   = lane >> 4;   // 0: K={0,1}, 1: K={2,3}
  const int kb   = hi * 2;

  for (int ct = wave; ct < NCOL / 16; ct += 8) {
    const int col = ct * 16 + lh;
    v8f acc = {};
    #pragma unroll 4
    for (int k0 = 0; k0 < K; k0 += 4) {           // mean @ Wl
      v2f a = *(const v2f*)&Am[lh][k0 + kb];
      v2f b;
      b.x = Wl[(size_t)(k0 + kb) * NCOL + col];
      b.y = Wl[(size_t)(k0 + kb + 1) * NCOL + col];
      acc = __builtin_amdgcn_wmma_f32_16x16x4_f32(false, a, false, b,
                                                  (short)0, acc, false, false);
    }
    #pragma unroll 4
    for (int k0 = 0; k0 < K; k0 += 4) {           // x @ Wr
      v2f a = *(const v2f*)&Ax[lh][k0 + kb];
      v2f b;
      b.x = Wr[(size_t)(k0 + kb) * NCOL + col];
      b.y = Wr[(size_t)(k0 + kb + 1) * NCOL + col];
      acc = __builtin_amdgcn_wmma_f32_16x16x4_f32(false, a, false, b,
                                                  (short)0, acc, false, false);
    }
    const float bv = bias[col];
    #pragma unroll
    for (int g = 0; g < 8; ++g) {                 // C layout: VGPR g -> row g+8*hi
      float v = acc[g] + bv;
      if (RELU) v = v > 0.0f ? v : 0.0f;
      out[(size_t)(row0 + g + 8 * hi) * NCOL + col] = v;
    }
  }
}

// ------------------------------------------------------------------ launch
extern "C" void kernel_launch(void* const* d_in, const int* in_sizes, int n_in,
                              void* d_out, int out_size, void* d_ws, size_t ws_size,
                              hipStream_t stream) {
  const float* x   = (const float*)d_in[0];
  const int*   ei  = (const int*)d_in[1];   // [2, E] int32
  const float* W1l = (const float*)d_in[2];
  const float* W1r = (const float*)d_in[3];
  const float* b1  = (const float*)d_in[4];
  const float* W2l = (const float*)d_in[5];
  const float* W2r = (const float*)d_in[6];
  const float* b2  = (const float*)d_in[7];
  float* out = (float*)d_out;

  float* ws = (float*)d_ws;
  const size_t DEG_PAD = 50176;  // >= N, multiple of 4
  float* deg  = ws;
  float* agg1 = deg + DEG_PAD;                         // N x F_IN
  float* agg2 = agg1 + (size_t)N_NODES * F_IN;         // N x F_HID
  float* h    = agg2 + (size_t)N_NODES * F_HID;        // N x F_HID

  // must re-zero accumulators every call (atomic accumulation, graph replay)
  long zeroChunks = ((long)DEG_PAD + (long)N_NODES * F_IN + (long)N_NODES * F_HID) / 4;
  zero_f32<<<8192, 256, 0, stream>>>((float4*)ws, zeroChunks);

  // layer 1: mean-aggregate x, then relu(mean@W1l + x@W1r + b1) -> h
  scatter_accum<F_IN, true><<<(N_EDGES * (F_IN / 4)) / 256, 256, 0, stream>>>(x, ei, agg1, deg);
  sage_gemm<F_IN, F_HID, true><<<N_NODES / 16, 256, 0, stream>>>(agg1, deg, x, W1l, W1r, b1, h);

  // layer 2: mean-aggregate h, then mean@W2l + h@W2r + b2 -> out
  scatter_accum<F_HID, false><<<(N_EDGES * (F_HID / 4)) / 256, 256, 0, stream>>>(h, ei, agg2, nullptr);
  sage_gemm<F_HID, F_OUT, false><<<N_NODES / 16, 256, 0, stream>>>(agg2, deg, h, W2l, W2r, b2, out);
}